// EnhancedHockeyGNN_89043261981338
// MI455X (gfx1250) — compile-verified
//
#include <hip/hip_runtime.h>
#include <hip/hip_bf16.h>
#include <math.h>

// ---------------------------------------------------------------------------
// EnhancedHockeyGNN forward on MI455X (gfx1250, wave32)
//
// Pipeline (all fp32, matching reference precision):
//   deg/dinv  -> xw1 = x@W1 (WMMA f32 16x16x4) -> edge scatter (atomic f32)
//   -> fused self-loop+bias+BN+ReLU -> xw2 = h1@W2 -> scatter -> fuse
//   -> head gather + 128x2 GEMV + log_softmax
// Bottleneck is the edge scatter (~1.6GB/layer); GEMMs are tiny (6.5 GFLOP).
// ---------------------------------------------------------------------------

typedef float v2f __attribute__((ext_vector_type(2)));
typedef float v8f __attribute__((ext_vector_type(8)));

#define HDIM 128
#define WT_STRIDE 132   // pad: (132n + k) % 64 banks distinct across 32 lanes
#define EPS_BN 1e-5f

__device__ __forceinline__ void atomic_add_f32(float* p, float v) {
    __hip_atomic_fetch_add(p, v, __ATOMIC_RELAXED, __HIP_MEMORY_SCOPE_AGENT);
}

// ---- utility: zero a buffer (float4 granularity) ---------------------------
__global__ __launch_bounds__(256) void k_zero(float4* p, long long n4) {
    long long i = (long long)blockIdx.x * blockDim.x + threadIdx.x;
    if (i < n4) p[i] = make_float4(0.f, 0.f, 0.f, 0.f);
}

// ---- degree accumulation ---------------------------------------------------
__global__ __launch_bounds__(256) void k_deg(const int* __restrict__ dst,
                                             float* __restrict__ deg, int nE) {
    int e = blockIdx.x * blockDim.x + threadIdx.x;
    if (e < nE) atomic_add_f32(&deg[dst[e]], 1.0f);
}

// ---- dinv = rsqrt(deg + 1)  (in place) -------------------------------------
__global__ __launch_bounds__(256) void k_dinv(float* __restrict__ deg, int n) {
    int i = blockIdx.x * blockDim.x + threadIdx.x;
    if (i < n) deg[i] = rsqrtf(deg[i] + 1.0f);
}

// ---- Y[N,128] = X[N,128] @ W[128,128]  via V_WMMA_F32_16X16X4_F32 ----------
// Block: 256 threads = 8 waves. Block stripe = 16 rows x 128 cols.
// Wave w -> 16x16 tile at columns [16w, 16w+16). K loop: 32 wmma of K=4.
__global__ __launch_bounds__(256) void k_gemm128_wmma(
        const float* __restrict__ X, const float* __restrict__ W,
        float* __restrict__ Y, int nrows) {
    extern __shared__ float wt[];                    // [128][WT_STRIDE] = W^T
    for (int i = threadIdx.x; i < HDIM * HDIM; i += 256) {
        int k = i >> 7, n = i & 127;
        wt[n * WT_STRIDE + k] = W[i];                // coalesced read, transpose
    }
    __syncthreads();

    const int lane = threadIdx.x & 31;
    const int wv   = threadIdx.x >> 5;               // 0..7
    const int m    = lane & 15;
    const int klo  = (lane >= 16) ? 2 : 0;           // ISA A/B half-wave K offset
    const int n    = wv * 16 + m;

    int row = blockIdx.x * 16 + m;
    if (row > nrows - 1) row = nrows - 1;            // keep EXEC all-1s for WMMA
    const float* xrow = X + (size_t)row * HDIM + klo;
    const v2f*   bptr = (const v2f*)(wt + n * WT_STRIDE + klo);

    v8f acc = {};
#pragma unroll
    for (int kk = 0; kk < 32; ++kk) {
        v2f a = *(const v2f*)(xrow + kk * 4);        // X[row][4kk+klo .. +1]
        v2f b = bptr[kk * 2];                        // W[4kk+klo..+1][n]
        acc = __builtin_amdgcn_wmma_f32_16x16x4_f32(
                  false, a, false, b, (short)0, acc, false, false);
    }

    // C layout: VGPR r -> M = r (lanes 0-15) / 8+r (lanes 16-31), N = lane&15
    const int r0 = blockIdx.x * 16 + ((lane >= 16) ? 8 : 0);
#pragma unroll
    for (int r = 0; r < 8; ++r) {
        int rr = r0 + r;
        if (rr < nrows) Y[(size_t)rr * HDIM + n] = acc[r];
    }
}

// ---- edge scatter: AGG[dst] += XW[src] * dinv[src]*dinv[dst] ---------------
// One wave per edge; lane handles 4 consecutive features (coalesced float4).
__global__ __launch_bounds__(256) void k_scatter(
        const float* __restrict__ XW, const float* __restrict__ dinv,
        const int* __restrict__ src, const int* __restrict__ dst,
        float* __restrict__ AGG, long long nE) {
    long long gid = (long long)blockIdx.x * blockDim.x + threadIdx.x;
    long long e = gid >> 5;
    if (e >= nE) return;
    int f = (int)(gid & 31) * 4;
    int s = src[e], d = dst[e];
    float nrm = dinv[s] * dinv[d];
    float4 v = *(const float4*)(XW + (size_t)s * HDIM + f);
    float* out = AGG + (size_t)d * HDIM + f;
    atomic_add_f32(out + 0, v.x * nrm);
    atomic_add_f32(out + 1, v.y * nrm);
    atomic_add_f32(out + 2, v.z * nrm);
    atomic_add_f32(out + 3, v.w * nrm);
}

// ---- fused epilogue: H = relu(BN(AGG + XW*dinv^2 + b))  (in place on AGG) --
__global__ __launch_bounds__(256) void k_epilogue(
        float* __restrict__ AGG, const float* __restrict__ XW,
        const float* __restrict__ dinv, const float* __restrict__ b,
        const float* __restrict__ gam, const float* __restrict__ bet,
        const float* __restrict__ mu,  const float* __restrict__ var,
        long long nQuads) {                           // nQuads = N*128/4
    long long i = (long long)blockIdx.x * blockDim.x + threadIdx.x;
    if (i >= nQuads) return;
    long long base = i * 4;
    int    c   = (int)(base & 127);
    size_t row = (size_t)(base >> 7);
    float di = dinv[row];
    float sl = di * di;
    float4 a = *(const float4*)(AGG + base);
    float4 x = *(const float4*)(XW + base);
    float4 bb = *(const float4*)(b + c);
    float4 g  = *(const float4*)(gam + c);
    float4 be = *(const float4*)(bet + c);
    float4 mm = *(const float4*)(mu + c);
    float4 vv = *(const float4*)(var + c);
    float4 o;
    o.x = fmaxf(0.f, (a.x + x.x * sl + bb.x - mm.x) * g.x * rsqrtf(vv.x + EPS_BN) + be.x);
    o.y = fmaxf(0.f, (a.y + x.y * sl + bb.y - mm.y) * g.y * rsqrtf(vv.y + EPS_BN) + be.y);
    o.z = fmaxf(0.f, (a.z + x.z * sl + bb.z - mm.z) * g.z * rsqrtf(vv.z + EPS_BN) + be.z);
    o.w = fmaxf(0.f, (a.w + x.w * sl + bb.w - mm.w) * g.w * rsqrtf(vv.w + EPS_BN) + be.w);
    *(float4*)(AGG + base) = o;
}

// ---- head: logits = H[idx] @ Wf + bf ; log_softmax over 2 classes ----------
// One wave per game; lane handles 4 features; shuffle reduction.
__global__ __launch_bounds__(256) void k_head(
        const float* __restrict__ H, const int* __restrict__ idx,
        const float* __restrict__ Wf, const float* __restrict__ bf,
        float* __restrict__ out, int nGames) {
    int g = blockIdx.x * (blockDim.x >> 5) + (threadIdx.x >> 5);
    int lane = threadIdx.x & 31;
    if (g >= nGames) return;
    int row = idx[g];
    float4 hv = *(const float4*)(H + (size_t)row * HDIM + lane * 4);
    float a0 = 0.f, a1 = 0.f;
    const float* wp = Wf + (size_t)(lane * 4) * 2;   // Wf is [128][2]
    a0 += hv.x * wp[0]; a1 += hv.x * wp[1];
    a0 += hv.y * wp[2]; a1 += hv.y * wp[3];
    a0 += hv.z * wp[4]; a1 += hv.z * wp[5];
    a0 += hv.w * wp[6]; a1 += hv.w * wp[7];
#pragma unroll
    for (int off = 16; off > 0; off >>= 1) {
        a0 += __shfl_xor(a0, off, 32);
        a1 += __shfl_xor(a1, off, 32);
    }
    if (lane == 0) {
        a0 += bf[0]; a1 += bf[1];
        float mx = fmaxf(a0, a1);
        float lse = logf(expf(a0 - mx) + expf(a1 - mx)) + mx;
        out[(size_t)g * 2 + 0] = a0 - lse;
        out[(size_t)g * 2 + 1] = a1 - lse;
    }
}

// ---------------------------------------------------------------------------
extern "C" void kernel_launch(void* const* d_in, const int* in_sizes, int n_in,
                              void* d_out, int out_size, void* d_ws, size_t ws_size,
                              hipStream_t stream) {
    const float* x   = (const float*)d_in[0];
    const int*   ei  = (const int*)d_in[1];       // [2,E] flat
    const int*   gix = (const int*)d_in[2];
    const float* W1  = (const float*)d_in[3];
    const float* b1  = (const float*)d_in[4];
    const float* g1  = (const float*)d_in[5];
    const float* be1 = (const float*)d_in[6];
    const float* m1  = (const float*)d_in[7];
    const float* v1  = (const float*)d_in[8];
    const float* W2  = (const float*)d_in[9];
    const float* b2  = (const float*)d_in[10];
    const float* g2  = (const float*)d_in[11];
    const float* be2 = (const float*)d_in[12];
    const float* m2  = (const float*)d_in[13];
    const float* v2  = (const float*)d_in[14];
    const float* Wf  = (const float*)d_in[15];
    const float* bf  = (const float*)d_in[16];

    const int  N_ = in_sizes[0] / HDIM;           // 100000
    const int  E_ = in_sizes[1] / 2;              // 1600000
    const int  NG = in_sizes[2];                  // 20000
    const int* src = ei;
    const int* dst = ei + E_;

    // workspace layout: dinv[N] | A[N*128] | B[N*128]
    float* dinv = (float*)d_ws;
    size_t nPad = ((size_t)N_ + 255) & ~(size_t)255;
    float* A = dinv + nPad;                       // xw buffer
    float* B = A + (size_t)N_ * HDIM;             // agg / h buffer

    const long long nhQ = (long long)N_ * HDIM / 4;   // float4 count of N*128
    const int TB = 256;
    dim3 blk(TB);
    auto g1d = [](long long n, int tb) { return dim3((unsigned)((n + tb - 1) / tb)); };

    const int gemmGrid = (N_ + 15) / 16;
    const size_t gemmLds = (size_t)HDIM * WT_STRIDE * sizeof(float);  // 67584 B
    const long long scatThreads = (long long)E_ * 32;

    // degrees -> dinv
    k_zero<<<g1d(nPad / 4, TB), blk, 0, stream>>>((float4*)dinv, (long long)nPad / 4);
    k_deg<<<g1d(E_, TB), blk, 0, stream>>>(dst, dinv, E_);
    k_dinv<<<g1d(N_, TB), blk, 0, stream>>>(dinv, N_);

    // ---- layer 1 ----
    k_gemm128_wmma<<<gemmGrid, blk, gemmLds, stream>>>(x, W1, A, N_);
    k_zero<<<g1d(nhQ, TB), blk, 0, stream>>>((float4*)B, nhQ);
    k_scatter<<<g1d(scatThreads, TB), blk, 0, stream>>>(A, dinv, src, dst, B, E_);
    k_epilogue<<<g1d(nhQ, TB), blk, 0, stream>>>(B, A, dinv, b1, g1, be1, m1, v1, nhQ);

    // ---- layer 2 ----
    k_gemm128_wmma<<<gemmGrid, blk, gemmLds, stream>>>(B, W2, A, N_);
    k_zero<<<g1d(nhQ, TB), blk, 0, stream>>>((float4*)B, nhQ);
    k_scatter<<<g1d(scatThreads, TB), blk, 0, stream>>>(A, dinv, src, dst, B, E_);
    k_epilogue<<<g1d(nhQ, TB), blk, 0, stream>>>(B, A, dinv, b2, g2, be2, m2, v2, nhQ);

    // ---- head ----
    k_head<<<g1d((long long)NG * 32, TB), blk, 0, stream>>>(B, gix, Wf, bf, (float*)d_out, NG);
}